// ConstraintLayer_33500744909471
// MI455X (gfx1250) — compile-verified
//
#include <hip/hip_runtime.h>

// ---------------------------------------------------------------------------
// ConstraintLayer on gfx1250 (MI455X, wave32, WMMA, TDM)
//
// base = aug0[:,0:256] x M^T  (M = 512 rows: 256 pos-mat rows then 256 neg),
// done with V_WMMA_F32_16X16X4_F32 (keeps full f32 precision).
// Then a cheap per-row sequential correction using the rank-1 update algebra:
//   dot_t(c) = base[t,c] + bias[t,c] + sum_{s<t} delta_s * M_t[c, vid_s]
// ---------------------------------------------------------------------------

typedef float v2f  __attribute__((ext_vector_type(2)));
typedef float v4f  __attribute__((ext_vector_type(4)));
typedef float v8f  __attribute__((ext_vector_type(8)));
typedef unsigned int u32x4 __attribute__((ext_vector_type(4)));
typedef int   i32x4 __attribute__((ext_vector_type(4)));
typedef int   i32x8 __attribute__((ext_vector_type(8)));

#ifndef __has_builtin
#define __has_builtin(x) 0
#endif

#if defined(__HIP_DEVICE_COMPILE__) && __has_builtin(__builtin_amdgcn_tensor_load_to_lds)
#define HAVE_TDM 1
#else
#define HAVE_TDM 0
#endif

constexpr int kN        = 256;   // variables (K of the GEMM)
constexpr int kC        = 8;     // constraints per matrix
constexpr int kV        = 32;    // constrained vars / scan steps
constexpr int ROWS      = 64;    // batch rows per block
constexpr int NTHREADS  = 128;   // 4 waves
constexpr int APITCH    = 260;   // padded LDS pitch (dwords) for 256-dword rows
constexpr int BPITCH    = 260;
constexpr int BASEPITCH = 68;    // padded pitch for 64-row base columns
constexpr int JCOLS     = 512;   // 2 * V * C GEMM output columns

constexpr int SMEM_FLOATS =
    ROWS * APITCH          // A tile (aug rows, bias col handled separately)
  + JCOLS * BASEPITCH      // base GEMM results, layout [j][row]
  + 2 * 16 * BPITCH        // double-buffered B tile
  + 2 * kV * kV * kC       // crossP/crossN [t][s][c]
  + 2 * kV * kC            // biasP/biasN
  + ROWS * kV;             // per-row delta history
constexpr int SMEM_BYTES = SMEM_FLOATS * 4 + kV * 4;  // + vids

// ---------------------------------------------------------------------------
// TDM: 2D tile (tile_k dwords per row, tile_rows rows, row_stride dwords in
// memory) -> LDS at byte offset lds_off, padding +4 dwords after every 256
// dwords so LDS rows land on a 260-dword pitch (bank-conflict free).
// ---------------------------------------------------------------------------
#if HAVE_TDM
__device__ __forceinline__ void tdm_load_tile(unsigned lds_off, const void* gptr,
                                              unsigned tile_k, unsigned tile_rows,
                                              unsigned row_stride) {
  unsigned long long ga = (unsigned long long)(size_t)gptr;
  u32x4 g0;
  g0.x = 1u;                                   // count=1, user descriptor
  g0.y = lds_off;                              // lds_addr [63:32]
  g0.z = (unsigned)ga;                         // global_addr [95:64]
  g0.w = ((unsigned)(ga >> 32) & 0x01FFFFFFu) | (2u << 30);  // addr hi | type=2
  i32x8 g1;
  g1.s0 = (int)((2u << 16) | (1u << 20) | (7u << 22) | (3u << 25)); // 4B, pad 4dw/256dw
  g1.s1 = (int)(tile_k << 16);                          // tensor_dim0 lo16
  g1.s2 = (int)((tile_k >> 16) | (tile_rows << 16));    // dim0 hi | dim1 lo
  g1.s3 = (int)((tile_rows >> 16) | (tile_k << 16));    // dim1 hi | tile_dim0
  g1.s4 = (int)tile_rows;                               // tile_dim1 (tile_dim2=0)
  g1.s5 = (int)row_stride;                              // tensor_dim0_stride lo32
  g1.s6 = 0;
  g1.s7 = 0;
  i32x4 z = {0, 0, 0, 0};
#if __has_include(<hip/amd_detail/amd_gfx1250_TDM.h>)
  i32x8 z8 = {0, 0, 0, 0, 0, 0, 0, 0};
  __builtin_amdgcn_tensor_load_to_lds(g0, g1, z, z, z8, 0);   // clang-23 6-arg form
#else
  __builtin_amdgcn_tensor_load_to_lds(g0, g1, z, z, 0);       // ROCm 7.2 5-arg form
#endif
}
__device__ __forceinline__ unsigned lds_off(const void* p) {
  return (unsigned)(size_t)p;   // flat LDS addr low 32 bits == LDS byte offset
}
#endif

__global__ __launch_bounds__(NTHREADS, 1)
void constraint_layer_fused(const float* __restrict__ preds,
                            const float* __restrict__ posm,
                            const float* __restrict__ negm,
                            const int*   __restrict__ var_ids,
                            float*       __restrict__ out) {
  extern __shared__ char smem_raw[];
  float* ldsA   = (float*)smem_raw;                 // [64][260]
  float* baseL  = ldsA   + ROWS * APITCH;           // [512][68]
  float* btile  = baseL  + JCOLS * BASEPITCH;       // [2][16][260]
  float* crossP = btile  + 2 * 16 * BPITCH;         // [32][32][8]
  float* crossN = crossP + kV * kV * kC;
  float* biasP  = crossN + kV * kV * kC;            // [32][8]
  float* biasN  = biasP  + kV * kC;
  float* deltaL = biasN  + kV * kC;                 // [64][32]
  int*   vlds   = (int*)(deltaL + ROWS * kV);       // [32]

  const int tid    = threadIdx.x;
  const int wave   = tid >> 5;        // 0..3
  const int lane   = tid & 31;
  const int l16    = lane & 15;
  const int hi     = lane >> 4;       // 0 = pos half, 1 = neg half
  const int rowBase = blockIdx.x * ROWS;

  // ---- stage A tile (and first B tile) via TDM, overlap with table gathers ----
#if HAVE_TDM
  if (wave == 0) {
    tdm_load_tile(lds_off(ldsA), preds + (size_t)rowBase * kN, kN, ROWS, kN);
    tdm_load_tile(lds_off(btile), posm, kN, 16, kN + 1);       // jt = 0
  }
#else
  for (int i = tid; i < ROWS * kN; i += NTHREADS)
    ldsA[(i >> 8) * APITCH + (i & 255)] = preds[(size_t)rowBase * kN + i];
#endif

  // cross tables [t][s][c] = mats[t][c][vid_s]; bias [t][c] = mats[t][c][256]
  for (int i = tid; i < kV * kV * kC; i += NTHREADS) {
    int t = i >> 8, s = (i >> 3) & 31, c = i & 7;
    int col = var_ids[s];
    int mrow = (t * kC + c) * (kN + 1);
    crossP[i] = posm[mrow + col];
    crossN[i] = negm[mrow + col];
  }
  for (int i = tid; i < kV * kC; i += NTHREADS) {
    biasP[i] = posm[i * (kN + 1) + kN];
    biasN[i] = negm[i * (kN + 1) + kN];
  }
  if (tid < kV) vlds[tid] = var_ids[tid];

  // ---- Phase 1: base GEMM, f32 WMMA 16x16x4, K = 256 ----
  const float* aRow = ldsA + (wave * 16 + l16) * APITCH + hi * 2;
  for (int jt = 0; jt < 32; ++jt) {
#if HAVE_TDM
    if (wave == 0) {
      if (jt + 1 < 32) {
        const float* src = (jt + 1 < 16) ? posm + (jt + 1) * 16 * (kN + 1)
                                         : negm + (jt + 1 - 16) * 16 * (kN + 1);
        tdm_load_tile(lds_off(btile + ((jt + 1) & 1) * 16 * BPITCH), src, kN, 16, kN + 1);
        __builtin_amdgcn_s_wait_tensorcnt(1);   // current tile done, next in flight
      } else {
        __builtin_amdgcn_s_wait_tensorcnt(0);
      }
    }
    __syncthreads();
    const float* bt = btile + (jt & 1) * 16 * BPITCH;
#else
    __syncthreads();
    const float* src = (jt < 16) ? posm + jt * 16 * (kN + 1)
                                 : negm + (jt - 16) * 16 * (kN + 1);
    for (int i = tid; i < 16 * kN; i += NTHREADS)
      btile[(i >> 8) * BPITCH + (i & 255)] = src[(i >> 8) * (kN + 1) + (i & 255)];
    __syncthreads();
    const float* bt = btile;
#endif

    v8f acc = {};
    const float* bRow = bt + l16 * BPITCH + hi * 2;
#pragma unroll 8
    for (int k = 0; k < 64; ++k) {
      v2f a = *(const v2f*)(aRow + 4 * k);   // A frag: rows, K = 4k + 2*hi + {0,1}
      v2f b = *(const v2f*)(bRow + 4 * k);   // B frag: cols, same K slots
      acc = __builtin_amdgcn_wmma_f32_16x16x4_f32(false, a, false, b,
                                                  (short)0, acc, false, false);
    }
    // store D: lane holds col j = jt*16 + l16, rows 8*hi .. 8*hi+7 of wave tile
    float* dst = baseL + (jt * 16 + l16) * BASEPITCH + wave * 16 + hi * 8;
    *(v4f*)dst       = (v4f){acc.s0, acc.s1, acc.s2, acc.s3};
    *(v4f*)(dst + 4) = (v4f){acc.s4, acc.s5, acc.s6, acc.s7};
    __syncthreads();   // protect B buffer before next TDM overwrites it
  }

  // ---- Phase 2: sequential 32-step correction, 2 lanes per row ----
  const int r = wave * 16 + l16;                   // row within block tile
  float* aRowP = ldsA + r * APITCH;
  float* drow  = deltaL + r * kV;
  const float* crossT0 = hi ? crossN : crossP;
  const float* biasT0  = hi ? biasN  : biasP;
  const float* baseHalf = baseL + (hi ? 256 : 0) * BASEPITCH;

  for (int t = 0; t < kV; ++t) {
    const int vid = vlds[t];
    const float curv = aRowP[vid];
    float acc[8];
    const float* bb = baseHalf + (t * 8) * BASEPITCH + r;
    const float* bs = biasT0 + t * 8;
#pragma unroll
    for (int c = 0; c < 8; ++c) acc[c] = bb[c * BASEPITCH] + bs[c];
    const float* ct = crossT0 + t * (kV * kC);
    for (int s = 0; s < t; ++s) {
      const float d = drow[s];
      v4f c0 = *(const v4f*)(ct + s * 8);
      v4f c1 = *(const v4f*)(ct + s * 8 + 4);
      acc[0] += d * c0.x; acc[1] += d * c0.y; acc[2] += d * c0.z; acc[3] += d * c0.w;
      acc[4] += d * c1.x; acc[5] += d * c1.y; acc[6] += d * c1.z; acc[7] += d * c1.w;
    }
    float red = hi ? 3.402823466e38f : -3.402823466e38f;
#pragma unroll
    for (int c = 0; c < 8; ++c) red = hi ? fminf(red, acc[c]) : fmaxf(red, acc[c]);
    const float other = __shfl_xor(red, 16, 32);
    const float pos_v = hi ? other : red;
    const float neg_v = hi ? red : other;
    const float newv  = fminf(fmaxf(curv, pos_v), neg_v);
    if (!hi) {                       // single writer per row; LDS is in-order per wave
      aRowP[vid] = newv;
      drow[t]    = newv - curv;
    }
  }

  // ---- write back: wave's 16 rows x 256 cols as float4 ----
#pragma unroll 4
  for (int i = 0; i < 32; ++i) {
    int q  = lane + 32 * i;          // 0..1023 float4 units in wave tile
    int rr = q >> 6;                 // row 0..15
    int c4 = q & 63;                 // float4 column
    v4f v = *(const v4f*)(ldsA + (wave * 16 + rr) * APITCH + c4 * 4);
    *(v4f*)(out + (size_t)(rowBase + wave * 16 + rr) * kN + c4 * 4) = v;
  }
}

extern "C" void kernel_launch(void* const* d_in, const int* in_sizes, int n_in,
                              void* d_out, int out_size, void* d_ws, size_t ws_size,
                              hipStream_t stream) {
  const float* preds = (const float*)d_in[0];
  const float* posm  = (const float*)d_in[1];
  const float* negm  = (const float*)d_in[2];
  const int*   vids  = (const int*)d_in[3];
  float* out = (float*)d_out;

  // opt in to >64KB dynamic LDS (deterministic; harmless to repeat per call)
  (void)hipFuncSetAttribute((const void*)constraint_layer_fused,
                            hipFuncAttributeMaxDynamicSharedMemorySize, SMEM_BYTES);

  const int blocks = 65536 / ROWS;   // 1024
  constraint_layer_fused<<<blocks, NTHREADS, SMEM_BYTES, stream>>>(
      preds, posm, negm, vids, out);
}